// SegHead_91079076479049
// MI455X (gfx1250) — compile-verified
//
#include <hip/hip_runtime.h>

// Problem constants from the reference (fixed shapes).
constexpr int B = 16, C = 128, N = 32768, S = 100;
constexpr int CHUNKS  = 32;           // chunks per batch -> 512 blocks total
constexpr int NCHUNK  = N / CHUNKS;   // 1024 points per block
constexpr int THREADS = 256;          // 8 wave32 waves
constexpr int NWAVES  = THREADS / 32;

#define NEG_INF (-__builtin_inff())

// CDNA5 async global->LDS copy path (builtin confirmed present by round-1 diag).
#if defined(__gfx1250__) && __has_builtin(__builtin_amdgcn_global_load_async_to_lds_b128) && \
    __has_builtin(__builtin_amdgcn_s_wait_asynccnt)
#define USE_ASYNC_LDS 1
#else
#define USE_ASYNC_LDS 0
#endif

typedef int   v4i __attribute__((ext_vector_type(4)));
typedef float v4f __attribute__((ext_vector_type(4)));

// Main kernel first so the disasm snippet shows its prologue (async copy + waits).
__global__ __launch_bounds__(THREADS) void seg_max_kernel(const float* __restrict__ feat,
                                                          const int* __restrict__ cid,
                                                          float* __restrict__ out) {
  // 51.2 KB accumulator, [c][s] layout so lanes (varying s, fixed c) spread banks.
  __shared__ float acc[C * S];
  __shared__ int   scid[NCHUNK];  // this chunk's cluster labels (4 KB)

  const int tid   = threadIdx.x;
  const int lane  = tid & 31;
  const int wid   = tid >> 5;
  const int b     = blockIdx.x / CHUNKS;
  const int chunk = blockIdx.x % CHUNKS;
  const int n0    = chunk * NCHUNK;

  // ---- 1) Stage cluster ids for this chunk into LDS (once per block). ----
  const int* gsrc = cid + (size_t)b * N + n0;
#if USE_ASYNC_LDS
  {
    typedef __attribute__((address_space(1))) v4i* gvec_t;
    typedef __attribute__((address_space(3))) v4i* lvec_t;
    // 256 threads x 16B = 4096B = NCHUNK ints, one async b128 per lane.
    gvec_t g = (gvec_t)(gsrc + tid * 4);
    lvec_t l = (lvec_t)(&scid[tid * 4]);
    __builtin_amdgcn_global_load_async_to_lds_b128(g, l, 0, 0);
  }
#else
  ((v4i*)scid)[tid] = ((const v4i*)gsrc)[tid];
#endif

  // ---- 2) Init accumulator to -inf (overlaps with the async copy). ----
  for (int i = tid; i < C * S; i += THREADS) acc[i] = NEG_INF;

#if USE_ASYNC_LDS
  __builtin_amdgcn_s_wait_asynccnt(0);
#endif
  __syncthreads();

  // ---- 3) Stream features; scatter-max into LDS via ds fp32 max atomics. ----
  // Each wave owns channel rows c = wid, wid+8, ...; lanes cover consecutive
  // points with b128 loads (coalesced 512B per wave instruction). The feature
  // stream (256 MB) is single-use and exceeds the 192 MB L2 -> non-temporal
  // loads keep L2 free for the output atomic traffic.
  const float* featb = feat + (size_t)b * C * N + n0;
  const v4i*   ip    = (const v4i*)scid;
  for (int c = wid; c < C; c += NWAVES) {
    const v4f* fp = (const v4f*)(featb + (size_t)c * N);
    if (c + NWAVES < C)  // prefetch next channel row (global_prefetch_b8)
      __builtin_prefetch(featb + (size_t)(c + NWAVES) * N + lane * 4, 0, 0);
    float* accrow = acc + c * S;
#pragma unroll
    for (int t = 0; t < NCHUNK / (32 * 4); ++t) {
      const int idx = t * 32 + lane;
      v4f f  = __builtin_nontemporal_load(&fp[idx]);
      v4i s4 = ip[idx];
      __hip_atomic_fetch_max(&accrow[s4.x], f.x, __ATOMIC_RELAXED, __HIP_MEMORY_SCOPE_WORKGROUP);
      __hip_atomic_fetch_max(&accrow[s4.y], f.y, __ATOMIC_RELAXED, __HIP_MEMORY_SCOPE_WORKGROUP);
      __hip_atomic_fetch_max(&accrow[s4.z], f.z, __ATOMIC_RELAXED, __HIP_MEMORY_SCOPE_WORKGROUP);
      __hip_atomic_fetch_max(&accrow[s4.w], f.w, __ATOMIC_RELAXED, __HIP_MEMORY_SCOPE_WORKGROUP);
    }
  }
  __syncthreads();

  // ---- 4) Flush block-local maxima to global with fp32 atomic max. ----
  // Iterate i = s*C + c so lanes hit consecutive output addresses; the LDS
  // read acc[c*S+s] then strides 100 words -> at most 2-way bank conflict.
  float* outb = out + (size_t)b * S * C;
  for (int i = tid; i < S * C; i += THREADS) {
    const int s_ = i >> 7;        // i / C
    const int c_ = i & (C - 1);   // i % C
    const float v = acc[c_ * S + s_];
    __hip_atomic_fetch_max(&outb[i], v, __ATOMIC_RELAXED, __HIP_MEMORY_SCOPE_AGENT);
  }
}

__global__ __launch_bounds__(THREADS) void seg_init_kernel(float* __restrict__ out, int n) {
  int i = blockIdx.x * THREADS + threadIdx.x;
  if (i < n) out[i] = NEG_INF;  // segment_max identity (empty segments -> -inf)
}

extern "C" void kernel_launch(void* const* d_in, const int* in_sizes, int n_in,
                              void* d_out, int out_size, void* d_ws, size_t ws_size,
                              hipStream_t stream) {
  (void)in_sizes; (void)n_in; (void)d_ws; (void)ws_size;
  const float* feat = (const float*)d_in[0];  // (B, C, N) fp32
  const int*   cid  = (const int*)d_in[1];    // (B, N) int32
  float*       out  = (float*)d_out;          // (B*S, C) fp32

  const int n = out_size;  // B*S*C = 204800
  seg_init_kernel<<<(n + THREADS - 1) / THREADS, THREADS, 0, stream>>>(out, n);
  seg_max_kernel<<<B * CHUNKS, THREADS, 0, stream>>>(feat, cid, out);
}